// Coat_quantize_bgn_64261300683329
// MI455X (gfx1250) — compile-verified
//
#include <hip/hip_runtime.h>

// COAT quantize-begin: per-(1x16) group amax -> scale -> e4m3 quantize.
// One lane owns one full 16-element group: zero cross-lane ops, coalesced
// per-lane Oscale store. 128-bit NT loads/stores; gfx1250 hardware FP8
// convert (v_cvt_pk_fp8_f32 / v_cvt_pk_f32_fp8) with RNE for the e4m3 grid.

typedef __attribute__((ext_vector_type(2))) float v2f;
typedef __attribute__((ext_vector_type(4))) float v4f;

#define E4M3_MAX 448.0f
#define INV_E4M3_MAX (1.0f / 448.0f)

__device__ __forceinline__ v4f quantize4(v4f v, float inv) {
    // Scale into e4m3 range, pack 4 f32 -> 4 fp8 bytes (RNE), re-expand to f32.
    float q0 = v.x * inv, q1 = v.y * inv, q2 = v.z * inv, q3 = v.w * inv;
    int packed = 0;
    packed = __builtin_amdgcn_cvt_pk_fp8_f32(q0, q1, packed, false); // bytes 0..1
    packed = __builtin_amdgcn_cvt_pk_fp8_f32(q2, q3, packed, true);  // bytes 2..3
    v2f lo = __builtin_amdgcn_cvt_pk_f32_fp8(packed, false);
    v2f hi = __builtin_amdgcn_cvt_pk_f32_fp8(packed, true);
    v4f r; r.x = lo.x; r.y = lo.y; r.z = hi.x; r.w = hi.y;
    return r;
}

__global__ __launch_bounds__(256) void coat_quantize_bgn_kernel(
    const float* __restrict__ x,
    float* __restrict__ out_x,
    float* __restrict__ out_q,
    float* __restrict__ out_s,
    unsigned int n_groups)                       // one group = 16 f32 = 64 B
{
    unsigned int t = blockIdx.x * blockDim.x + threadIdx.x;
    if (t >= n_groups) return;

    const v4f* xin = reinterpret_cast<const v4f*>(x) + (size_t)t * 4u;
    v4f* xo = reinterpret_cast<v4f*>(out_x) + (size_t)t * 4u;
    v4f* qo = reinterpret_cast<v4f*>(out_q) + (size_t)t * 4u;

    // 4x 128-bit NT loads: this lane's whole 16-element group (64 B).
    v4f a0 = __builtin_nontemporal_load(xin + 0);
    v4f a1 = __builtin_nontemporal_load(xin + 1);
    v4f a2 = __builtin_nontemporal_load(xin + 2);
    v4f a3 = __builtin_nontemporal_load(xin + 3);

    // In-register absmax over 16 values (lowers to v_max3 with |..| modifiers).
    float m01 = fmaxf(fmaxf(fabsf(a0.x), fabsf(a0.y)), fmaxf(fabsf(a0.z), fabsf(a0.w)));
    m01 = fmaxf(m01, fmaxf(fmaxf(fabsf(a1.x), fabsf(a1.y)), fmaxf(fabsf(a1.z), fabsf(a1.w))));
    float m23 = fmaxf(fmaxf(fabsf(a2.x), fabsf(a2.y)), fmaxf(fabsf(a2.z), fabsf(a2.w)));
    m23 = fmaxf(m23, fmaxf(fmaxf(fabsf(a3.x), fabsf(a3.y)), fmaxf(fabsf(a3.z), fabsf(a3.w))));
    float amax = fmaxf(m01, m23);

    // scale = amax/448 (all-zero group -> 1.0). inv via v_rcp_f32: 1-ulp error,
    // far below the fp8 RNE tie granularity (3-bit mantissa).
    bool zero = (amax == 0.0f);
    float scale = zero ? 1.0f : amax * INV_E4M3_MAX;
    float inv   = zero ? 1.0f : E4M3_MAX * __builtin_amdgcn_rcpf(amax);

    // Passthrough copy of x (streaming NT stores).
    __builtin_nontemporal_store(a0, xo + 0);
    __builtin_nontemporal_store(a1, xo + 1);
    __builtin_nontemporal_store(a2, xo + 2);
    __builtin_nontemporal_store(a3, xo + 3);

    // Q = fp8-grid value of v/scale (reference does NOT rescale), f32 view.
    __builtin_nontemporal_store(quantize4(a0, inv), qo + 0);
    __builtin_nontemporal_store(quantize4(a1, inv), qo + 1);
    __builtin_nontemporal_store(quantize4(a2, inv), qo + 2);
    __builtin_nontemporal_store(quantize4(a3, inv), qo + 3);

    // One scale per group, one lane per group: fully coalesced b32 store.
    __builtin_nontemporal_store(scale, out_s + t);
}

extern "C" void kernel_launch(void* const* d_in, const int* in_sizes, int n_in,
                              void* d_out, int out_size, void* d_ws, size_t ws_size,
                              hipStream_t stream) {
    (void)n_in; (void)d_ws; (void)ws_size; (void)out_size;

    const float* x = (const float*)d_in[0];
    // d_in[1] is group_size (==16); layout below hard-assumes 16.

    const unsigned long long n = (unsigned long long)in_sizes[0]; // 4*4096*4096

    float* out_x = (float*)d_out;            // passthrough x        [n]
    float* out_q = out_x + n;                // Q (f32 view of fp8)  [n]
    float* out_s = out_q + n;                // Oscale               [n/16]

    const unsigned int n_groups = (unsigned int)(n / 16ull);   // 4,194,304
    const unsigned int block = 256;
    const unsigned int grid = (n_groups + block - 1) / block;  // 16384

    coat_quantize_bgn_kernel<<<grid, block, 0, stream>>>(x, out_x, out_q, out_s, n_groups);
}